// GIN_25331717112176
// MI455X (gfx1250) — compile-verified
//
#include <hip/hip_runtime.h>

// ---------------------------------------------------------------------------
// GIN (2x GINConv + BN + ReLU, max/mean pool, FC) for MI455X / gfx1250.
//
// Memory-bound: edge list (128 MB HBM) dominates; features (16 MB) live in L2.
// MLPs use V_WMMA_F32_16X16X4_F32 (f32 A/B/C -> exact f32 math, wave32,
// 16-node tiles per wave), chained over K in steps of 4.
// ---------------------------------------------------------------------------

#define N_NODES  250000
#define N_EDGES  8000000
#define N_GRAPHS 512
#define BN_EPS   1e-5f

#define N_TILES  (N_NODES / 16)      // 15625 exactly
#define WAVES_PER_BLOCK 4

typedef float v2f __attribute__((ext_vector_type(2)));
typedef float v8f __attribute__((ext_vector_type(8)));

// ----------------------------- utility -------------------------------------
__global__ void k_zero_f32(float* __restrict__ p, int n) {
  int i = blockIdx.x * blockDim.x + threadIdx.x;
  if (i < n) p[i] = 0.0f;
}

// --------------------- edge scatter: agg[dst] += h[src] --------------------
__global__ void k_scatter(const float* __restrict__ h,
                          const long long* __restrict__ src,
                          const long long* __restrict__ dst,
                          float* __restrict__ agg) {
  int e = blockIdx.x * blockDim.x + threadIdx.x;
  if (e >= N_EDGES) return;
  int s = (int)src[e];
  int d = (int)dst[e];
  const float4* xs = (const float4*)(h + (size_t)s * 16);
  float* a = agg + (size_t)d * 16;
#pragma unroll
  for (int q = 0; q < 4; ++q) {
    float4 v = xs[q];                       // 16B gather (L2 resident)
    atomicAdd(a + q * 4 + 0, v.x);          // global_atomic_add_f32
    atomicAdd(a + q * 4 + 1, v.y);
    atomicAdd(a + q * 4 + 2, v.z);
    atomicAdd(a + q * 4 + 3, v.w);
  }
}

// --------------- layer-1 MLP: relu(relu((x+agg)W1a+b1a)W1b+b1b) ------------
// One wave = one 16-node tile. A-frag layout for 16x16x4 f32 WMMA:
//   lane L: m = L&15, khalf = (L>>4)*2; a = {A[m][kb], A[m][kb+1]}, kb=4s+khalf
// B-frag: lane L: n = L&15; b = {B[kb][n], B[kb+1][n]}
// C/D:    lane L, vgpr r: C[r + 8*(L>>4)][n]
__global__ void k_mlp1(const float* __restrict__ x, const float* __restrict__ agg,
                       const float* __restrict__ W1a, const float* __restrict__ b1a,
                       const float* __restrict__ W1b, const float* __restrict__ b1b,
                       float* __restrict__ u, float* __restrict__ sums /*[32]*/) {
  const int lane = threadIdx.x & 31;
  const int wv   = threadIdx.x >> 5;
  const int tile = blockIdx.x * WAVES_PER_BLOCK + wv;
  const bool active = (tile < N_TILES);            // wave-uniform
  const int base = tile * 16;

  __shared__ float ldsA[WAVES_PER_BLOCK][16 * 17]; // 16x16 in, pad 17
  __shared__ float ldsH[WAVES_PER_BLOCK][16 * 33]; // 16x32 mid, pad 33

  const int m    = lane & 15;
  const int hi   = lane >> 4;
  const int hi2  = hi * 2;
  const int n    = lane & 15;
  const int rowo = hi * 8;

  if (active) {   // load tile of (x + agg), row-major into LDS
    const size_t off = (size_t)(base + m) * 16 + hi * 8;
#pragma unroll
    for (int f = 0; f < 8; ++f)
      ldsA[wv][m * 17 + hi * 8 + f] = x[off + f] + agg[off + f];
  }
  __syncthreads();

  if (active) {   // matmul1: (16x16)@(16x32), K in 4 steps, two 16-col tiles
    v8f c0 = {}; v8f c1 = {};
#pragma unroll
    for (int s = 0; s < 4; ++s) {
      int kb = 4 * s + hi2;
      v2f a;  a.x  = ldsA[wv][m * 17 + kb];      a.y  = ldsA[wv][m * 17 + kb + 1];
      v2f b0; b0.x = W1a[kb * 32 + n];           b0.y = W1a[(kb + 1) * 32 + n];
      v2f b1; b1.x = W1a[kb * 32 + 16 + n];      b1.y = W1a[(kb + 1) * 32 + 16 + n];
      c0 = __builtin_amdgcn_wmma_f32_16x16x4_f32(false, a, false, b0, (short)0, c0, false, false);
      c1 = __builtin_amdgcn_wmma_f32_16x16x4_f32(false, a, false, b1, (short)0, c1, false, false);
    }
    float bia0 = b1a[n], bia1 = b1a[16 + n];
#pragma unroll
    for (int r = 0; r < 8; ++r) {                // bias+relu, D-layout -> LDS rows
      ldsH[wv][(r + rowo) * 33 + n]      = fmaxf(c0[r] + bia0, 0.0f);
      ldsH[wv][(r + rowo) * 33 + 16 + n] = fmaxf(c1[r] + bia1, 0.0f);
    }
  }
  __syncthreads();

  if (active) {   // matmul2: (16x32)@(32x16), K in 8 steps
    v8f c2 = {};
#pragma unroll
    for (int s = 0; s < 8; ++s) {
      int kb = 4 * s + hi2;
      v2f a; a.x = ldsH[wv][m * 33 + kb];  a.y = ldsH[wv][m * 33 + kb + 1];
      v2f b; b.x = W1b[kb * 16 + n];       b.y = W1b[(kb + 1) * 16 + n];
      c2 = __builtin_amdgcn_wmma_f32_16x16x4_f32(false, a, false, b, (short)0, c2, false, false);
    }
    float bib = b1b[n];
    float psum = 0.0f, psq = 0.0f;
#pragma unroll
    for (int r = 0; r < 8; ++r) {
      float v = fmaxf(c2[r] + bib, 0.0f);
      u[(size_t)(base + r + rowo) * 16 + n] = v;
      psum += v; psq += v * v;
    }
    psum += __shfl_xor(psum, 16);               // fold half-waves (same feature n)
    psq  += __shfl_xor(psq, 16);
    if (lane < 16) {
      atomicAdd(&sums[n],      psum);           // BN batch statistics
      atomicAdd(&sums[16 + n], psq);
    }
  }
}

// ---------------- layer-2 MLP: 16->16->16, same wave-tile scheme -----------
__global__ void k_mlp2(const float* __restrict__ h, const float* __restrict__ agg,
                       const float* __restrict__ W2a, const float* __restrict__ b2a,
                       const float* __restrict__ W2b, const float* __restrict__ b2b,
                       float* __restrict__ u, float* __restrict__ sums /*[32]*/) {
  const int lane = threadIdx.x & 31;
  const int wv   = threadIdx.x >> 5;
  const int tile = blockIdx.x * WAVES_PER_BLOCK + wv;
  const bool active = (tile < N_TILES);
  const int base = tile * 16;

  __shared__ float ldsA[WAVES_PER_BLOCK][16 * 17];
  __shared__ float ldsH[WAVES_PER_BLOCK][16 * 17];

  const int m    = lane & 15;
  const int hi   = lane >> 4;
  const int hi2  = hi * 2;
  const int n    = lane & 15;
  const int rowo = hi * 8;

  if (active) {
    const size_t off = (size_t)(base + m) * 16 + hi * 8;
#pragma unroll
    for (int f = 0; f < 8; ++f)
      ldsA[wv][m * 17 + hi * 8 + f] = h[off + f] + agg[off + f];
  }
  __syncthreads();

  if (active) {
    v8f c0 = {};
#pragma unroll
    for (int s = 0; s < 4; ++s) {
      int kb = 4 * s + hi2;
      v2f a; a.x = ldsA[wv][m * 17 + kb];  a.y = ldsA[wv][m * 17 + kb + 1];
      v2f b; b.x = W2a[kb * 16 + n];       b.y = W2a[(kb + 1) * 16 + n];
      c0 = __builtin_amdgcn_wmma_f32_16x16x4_f32(false, a, false, b, (short)0, c0, false, false);
    }
    float bia = b2a[n];
#pragma unroll
    for (int r = 0; r < 8; ++r)
      ldsH[wv][(r + rowo) * 17 + n] = fmaxf(c0[r] + bia, 0.0f);
  }
  __syncthreads();

  if (active) {
    v8f c2 = {};
#pragma unroll
    for (int s = 0; s < 4; ++s) {
      int kb = 4 * s + hi2;
      v2f a; a.x = ldsH[wv][m * 17 + kb];  a.y = ldsH[wv][m * 17 + kb + 1];
      v2f b; b.x = W2b[kb * 16 + n];       b.y = W2b[(kb + 1) * 16 + n];
      c2 = __builtin_amdgcn_wmma_f32_16x16x4_f32(false, a, false, b, (short)0, c2, false, false);
    }
    float bib = b2b[n];
    float psum = 0.0f, psq = 0.0f;
#pragma unroll
    for (int r = 0; r < 8; ++r) {
      float v = fmaxf(c2[r] + bib, 0.0f);
      u[(size_t)(base + r + rowo) * 16 + n] = v;
      psum += v; psq += v * v;
    }
    psum += __shfl_xor(psum, 16);
    psq  += __shfl_xor(psq, 16);
    if (lane < 16) {
      atomicAdd(&sums[n],      psum);
      atomicAdd(&sums[16 + n], psq);
    }
  }
}

// ------------- BN stats -> fused per-feature scale/shift (16 wide) ---------
__global__ void k_bnstats(const float* __restrict__ sums,
                          const float* __restrict__ gamma, const float* __restrict__ beta,
                          float* __restrict__ ss /*[32]: scale, shift*/) {
  int f = threadIdx.x;
  if (f < 16) {
    float mean = sums[f] * (1.0f / (float)N_NODES);
    float var  = sums[16 + f] * (1.0f / (float)N_NODES) - mean * mean;  // biased var
    float sc   = gamma[f] * rsqrtf(var + BN_EPS);
    ss[f]      = sc;
    ss[16 + f] = beta[f] - mean * sc;
  }
}

// ------------------------ BN apply + ReLU (elementwise) --------------------
__global__ void k_bnrelu(const float* __restrict__ u, const float* __restrict__ ss,
                         float* __restrict__ h) {
  int i = blockIdx.x * blockDim.x + threadIdx.x;
  if (i < N_NODES * 16) {
    int f = i & 15;
    h[i] = fmaxf(u[i] * ss[f] + ss[16 + f], 0.0f);
  }
}

// ------------- BN apply + ReLU fused with max/mean graph pooling -----------
// Post-ReLU values >= 0, so uint bit-ordering == float ordering for atomicMax,
// and the zero-initialized max buffer reproduces the isfinite->0 guard.
__global__ void k_bnrelu_pool(const float* __restrict__ u, const float* __restrict__ ss,
                              const long long* __restrict__ batch,
                              float* __restrict__ pool_max, float* __restrict__ pool_sum,
                              int* __restrict__ counts) {
  int i = blockIdx.x * blockDim.x + threadIdx.x;
  if (i >= N_NODES * 16) return;
  int node = i >> 4, f = i & 15;
  float v = fmaxf(u[i] * ss[f] + ss[16 + f], 0.0f);
  int g = (int)batch[node];
  atomicAdd(&pool_sum[g * 16 + f], v);
  atomicMax((unsigned int*)&pool_max[g * 16 + f], __float_as_uint(v));
  if (f == 0) atomicAdd(&counts[g], 1);
}

// --------------------------- final FC: [G,32]@[32,2] -----------------------
__global__ void k_fc(const float* __restrict__ pool_max, const float* __restrict__ pool_sum,
                     const int* __restrict__ counts,
                     const float* __restrict__ Wfc, const float* __restrict__ bfc,
                     float* __restrict__ out) {
  int t = blockIdx.x * blockDim.x + threadIdx.x;
  if (t >= N_GRAPHS * 2) return;
  int g = t >> 1, o = t & 1;
  float inv_cnt = 1.0f / fmaxf((float)counts[g], 1.0f);
  float acc = bfc[o];
#pragma unroll
  for (int k = 0; k < 16; ++k) acc += pool_max[g * 16 + k] * Wfc[k * 2 + o];
#pragma unroll
  for (int k = 0; k < 16; ++k) acc += (pool_sum[g * 16 + k] * inv_cnt) * Wfc[(16 + k) * 2 + o];
  out[t] = acc;
}

// ---------------------------------------------------------------------------
extern "C" void kernel_launch(void* const* d_in, const int* in_sizes, int n_in,
                              void* d_out, int out_size, void* d_ws, size_t ws_size,
                              hipStream_t stream) {
  const float* x   = (const float*)d_in[0];
  const float* W1a = (const float*)d_in[1];
  const float* b1a = (const float*)d_in[2];
  const float* W1b = (const float*)d_in[3];
  const float* b1b = (const float*)d_in[4];
  const float* g1  = (const float*)d_in[5];
  const float* be1 = (const float*)d_in[6];
  const float* W2a = (const float*)d_in[7];
  const float* b2a = (const float*)d_in[8];
  const float* W2b = (const float*)d_in[9];
  const float* b2b = (const float*)d_in[10];
  const float* g2  = (const float*)d_in[11];
  const float* be2 = (const float*)d_in[12];
  const float* Wfc = (const float*)d_in[13];
  const float* bfc = (const float*)d_in[14];
  const long long* ei    = (const long long*)d_in[15];  // int64 [2, N_EDGES]
  const long long* batch = (const long long*)d_in[16];  // int64 [N_NODES]
  const long long* srcv = ei;
  const long long* dstv = ei + N_EDGES;

  // workspace carve-up
  float* agg      = (float*)d_ws;            // 4,000,000 f32
  float* u        = agg + (size_t)N_NODES * 16;
  float* h        = u   + (size_t)N_NODES * 16;
  float* sums     = h   + (size_t)N_NODES * 16;   // [32] sum / sumsq
  float* ss       = sums + 32;                    // [32] scale / shift
  float* pool_max = ss + 32;                      // [512*16]
  float* pool_sum = pool_max + N_GRAPHS * 16;     // [512*16]
  int*   counts   = (int*)(pool_sum + N_GRAPHS * 16); // [512]

  const int NF = N_NODES * 16;
  const int mlp_blocks  = (N_TILES + WAVES_PER_BLOCK - 1) / WAVES_PER_BLOCK;
  const int ew_blocks   = (NF + 255) / 256;
  const int edge_blocks = (N_EDGES + 255) / 256;

  // ---- layer 1 ----
  k_zero_f32<<<ew_blocks, 256, 0, stream>>>(agg, NF);
  k_zero_f32<<<1, 64, 0, stream>>>(sums, 32);
  k_scatter<<<edge_blocks, 256, 0, stream>>>(x, srcv, dstv, agg);
  k_mlp1<<<mlp_blocks, 32 * WAVES_PER_BLOCK, 0, stream>>>(x, agg, W1a, b1a, W1b, b1b, u, sums);
  k_bnstats<<<1, 32, 0, stream>>>(sums, g1, be1, ss);
  k_bnrelu<<<ew_blocks, 256, 0, stream>>>(u, ss, h);

  // ---- layer 2 ----
  k_zero_f32<<<ew_blocks, 256, 0, stream>>>(agg, NF);
  k_zero_f32<<<1, 64, 0, stream>>>(sums, 32);
  k_scatter<<<edge_blocks, 256, 0, stream>>>(h, srcv, dstv, agg);
  k_mlp2<<<mlp_blocks, 32 * WAVES_PER_BLOCK, 0, stream>>>(h, agg, W2a, b2a, W2b, b2b, u, sums);
  k_bnstats<<<1, 32, 0, stream>>>(sums, g2, be2, ss);

  // ---- pooling + FC ----
  k_zero_f32<<<(2 * N_GRAPHS * 16 + N_GRAPHS + 255) / 256, 256, 0, stream>>>(pool_max,
                                                       2 * N_GRAPHS * 16 + N_GRAPHS);
  k_bnrelu_pool<<<ew_blocks, 256, 0, stream>>>(u, ss, batch, pool_max, pool_sum, counts);
  k_fc<<<(N_GRAPHS * 2 + 255) / 256, 256, 0, stream>>>(pool_max, pool_sum, counts, Wfc, bfc,
                                                       (float*)d_out);
}